// TemporalAugmentation_27298812134144
// MI455X (gfx1250) — compile-verified
//
#include <hip/hip_runtime.h>
#include <stdint.h>
#include <math.h>

// Problem constants (from reference)
#define B_      32
#define T_IN_   512
#define N_      543
#define C_      3
#define NC_     (N_ * C_)          // 1629
#define NEW_T_  426                 // int(512 / 1.2)
#define JF_     53                  // jitter freq
#define TILE_   32                  // channel tile (one lane per channel)
#define NTILES_ ((NC_ + TILE_ - 1) / TILE_)   // 51

typedef __attribute__((ext_vector_type(4))) unsigned int v4u;
typedef __attribute__((ext_vector_type(8))) int          v8i;
typedef __attribute__((ext_vector_type(4))) int          v4i;

#if defined(__gfx1250__) && __has_builtin(__builtin_amdgcn_tensor_load_to_lds)
#define HAVE_TDM 1
#else
#define HAVE_TDM 0
#endif

// ---------------------------------------------------------------------------
// Prep kernel: per-batch scalar work (kept-frame scan, gather indices, jitter
// resample, output mask). Tiny: 32 blocks x 256 threads.
// ---------------------------------------------------------------------------
__global__ void prep_kernel(const unsigned char* __restrict__ mask,
                            const unsigned char* __restrict__ keep,
                            const float* __restrict__ bj,
                            int* __restrict__ idx0, int* __restrict__ idx1,
                            float* __restrict__ frac, float* __restrict__ jit,
                            float* __restrict__ mask_out) {
  __shared__ int kept[NEW_T_];
  __shared__ int Ksh;
  const int b = blockIdx.x;
  if (threadIdx.x == 0) {
    int k = 0;
    for (int t = 0; t < NEW_T_; ++t)
      if (keep[b * NEW_T_ + t]) kept[k++] = t;
    Ksh = k;                       // first/last forced kept -> K >= 2
  }
  __syncthreads();
  const int K = Ksh;
  for (int t = threadIdx.x; t < NEW_T_; t += blockDim.x) {
    // frame-drop re-interp coordinates: s = t*(K-1)/(T2-1)
    float s = (float)t * (float)(K - 1) / (float)(NEW_T_ - 1);
    int r0 = (int)floorf(s);
    if (r0 < 0) r0 = 0;
    if (r0 > K - 2) r0 = K - 2;
    frac[b * NEW_T_ + t] = s - (float)r0;
    idx0[b * NEW_T_ + t] = kept[r0];
    idx1[b * NEW_T_ + t] = kept[r0 + 1];
    // jitter: linresample(base_jitter, 53 -> 426), * JITTER_STD
    float s2 = (float)t * (float)((double)(JF_ - 1) / (double)(NEW_T_ - 1));
    int j0 = (int)floorf(s2);
    if (j0 < 0) j0 = 0;
    if (j0 > JF_ - 2) j0 = JF_ - 2;
    float f2 = s2 - (float)j0;
    jit[b * NEW_T_ + t] =
        (bj[b * JF_ + j0] * (1.f - f2) + bj[b * JF_ + j0 + 1] * f2) * 0.02f;
    // mask: mask[b, floor(t * T/NEW_T)] & keep
    int nidx = (int)floorf((float)t * (float)((double)T_IN_ / (double)NEW_T_));
    if (nidx > T_IN_ - 1) nidx = T_IN_ - 1;
    mask_out[b * NEW_T_ + t] =
        (mask[b * T_IN_ + nidx] && keep[b * NEW_T_ + t]) ? 1.0f : 0.0f;
  }
}

// ---------------------------------------------------------------------------
// Main fused kernel. One block = (batch b, 32-channel tile). The raw time
// series x[b, 0:512, nc0:nc0+32] is staged in LDS (64KB) by the Tensor Data
// Mover (2-D strided tile descriptor); all resampling reads hit LDS.
// ---------------------------------------------------------------------------
__device__ __forceinline__ float x1_at(const float* sm, int t2, int ch) {
  // time resample, align_corners: s = t2 * (T_IN-1)/(NEW_T-1)
  float s = (float)t2 * (float)((double)(T_IN_ - 1) / (double)(NEW_T_ - 1));
  int i0 = (int)s;                 // s >= 0 -> trunc == floor
  if (i0 > T_IN_ - 2) i0 = T_IN_ - 2;
  float f = s - (float)i0;
  return sm[i0 * TILE_ + ch] * (1.f - f) + sm[(i0 + 1) * TILE_ + ch] * f;
}

__global__ __launch_bounds__(256) void main_kernel(
    const float* __restrict__ x, const unsigned char* __restrict__ keep,
    const float* __restrict__ noise, const float* __restrict__ spat,
    const int* __restrict__ idx0, const int* __restrict__ idx1,
    const float* __restrict__ frac, const float* __restrict__ jit,
    float* __restrict__ out) {
  __shared__ float sm[T_IN_ * TILE_];          // 64 KB raw tile
  const int tile = blockIdx.x;
  const int b    = blockIdx.y;
  const int nc0  = tile * TILE_;
  int valid_w = NC_ - nc0;
  if (valid_w > TILE_) valid_w = TILE_;

#if HAVE_TDM
  // --- TDM: DMA the strided 2-D tile (tile_dim0=32 elems, tile_dim1=512
  // rows, row stride = NC elements) from global into LDS. tensor_dim0 is the
  // remaining channel extent so the ragged last tile reads OOB as zeros.
  if (threadIdx.x < 32) {                      // one wave issues once
    uint64_t ga  = (uint64_t)(uintptr_t)(x + (size_t)b * T_IN_ * NC_ + nc0);
    uint32_t lds = (uint32_t)(uintptr_t)&sm[0];   // low 32 bits = LDS offset
    v4u g0;
    g0[0] = 1u;                                // count=1, user descriptor
    g0[1] = lds;                               // lds_addr
    g0[2] = (uint32_t)ga;                      // global_addr[31:0]
    g0[3] = (uint32_t)((ga >> 32) & 0x01FFFFFFu) | (2u << 30);  // addr hi | type=2
    v8i g1;
    g1[0] = (int)(2u << 16);                   // data_size = 2 (4 bytes)
    g1[1] = (int)(((uint32_t)valid_w & 0xFFFFu) << 16);          // tensor_dim0[15:0]
    g1[2] = (int)((((uint32_t)valid_w >> 16) & 0xFFFFu)
                  | ((uint32_t)T_IN_ << 16));  // tensor_dim0[31:16] | tensor_dim1[15:0]
    g1[3] = (int)((uint32_t)TILE_ << 16);      // tensor_dim1[31:16]=0 | tile_dim0
    g1[4] = T_IN_;                             // tile_dim1=512 | tile_dim2=0
    g1[5] = NC_;                               // tensor_dim0_stride[31:0]
    g1[6] = 0;                                 // stride hi | dim1_stride lo (unused, 2D)
    g1[7] = 0;
    v4i z4 = {0, 0, 0, 0};
    v8i z8 = {0, 0, 0, 0, 0, 0, 0, 0};
    __builtin_amdgcn_tensor_load_to_lds(g0, g1, z4, z4, z8, 0);
    __builtin_amdgcn_s_wait_tensorcnt(0);
  }
  __syncthreads();
#else
  // Fallback cooperative copy (keeps the source compiling on any toolchain).
  for (int i = threadIdx.x; i < T_IN_ * TILE_; i += blockDim.x) {
    int r = i >> 5, c = i & 31;
    sm[i] = (c < valid_w) ? x[((size_t)b * T_IN_ + r) * NC_ + nc0 + c] : 0.f;
  }
  __syncthreads();
#endif

  const int lane = threadIdx.x & 31;           // channel within tile
  const int wv   = threadIdx.x >> 5;           // 8 waves, one t each
  const int nc   = nc0 + lane;
  const bool act = lane < valid_w;

  const unsigned char* kb  = keep + (size_t)b * NEW_T_;
  const int*   i0b = idx0 + b * NEW_T_;
  const int*   i1b = idx1 + b * NEW_T_;
  const float* fb  = frac + b * NEW_T_;
  const float* jb  = jit  + b * NEW_T_;
  const float  sp  = act ? spat[(size_t)b * NC_ + nc] * 0.005f : 0.f;

  for (int t = wv; t < NEW_T_; t += 8) {       // wave-uniform t
    float x2t;
    if (kb[t]) {
      x2t = x1_at(sm, t, lane);
    } else {
      float f = fb[t];
      x2t = x1_at(sm, i0b[t], lane) * (1.f - f) + x1_at(sm, i1b[t], lane) * f;
    }
    float x3;
    if (t == 0) {
      x3 = x2t;
    } else {
      float x2p;
      if (kb[t - 1]) {
        x2p = x1_at(sm, t - 1, lane);
      } else {
        float f = fb[t - 1];
        x2p = x1_at(sm, i0b[t - 1], lane) * (1.f - f)
            + x1_at(sm, i1b[t - 1], lane) * f;
      }
      x3 = x2t + (x2t - x2p) * jb[t];          // velocity jitter
    }
    if (act) {
      size_t o = ((size_t)b * NEW_T_ + t) * NC_ + nc;
      out[o] = x3 + noise[o] * 0.01f + sp;
    }
  }
}

// ---------------------------------------------------------------------------
extern "C" void kernel_launch(void* const* d_in, const int* in_sizes, int n_in,
                              void* d_out, int out_size, void* d_ws,
                              size_t ws_size, hipStream_t stream) {
  const float*         x     = (const float*)d_in[0];
  const unsigned char* mask  = (const unsigned char*)d_in[1];  // bool
  const unsigned char* keep  = (const unsigned char*)d_in[2];  // bool
  const float*         bj    = (const float*)d_in[3];
  const float*         noise = (const float*)d_in[4];
  const float*         spat  = (const float*)d_in[5];

  float* out_x    = (float*)d_out;
  float* out_mask = out_x + (size_t)B_ * NEW_T_ * NC_;  // tuple: x then mask

  int*   idx0 = (int*)d_ws;
  int*   idx1 = idx0 + B_ * NEW_T_;
  float* frac = (float*)(idx1 + B_ * NEW_T_);
  float* jit  = frac + B_ * NEW_T_;

  prep_kernel<<<B_, 256, 0, stream>>>(mask, keep, bj, idx0, idx1, frac, jit,
                                      out_mask);
  dim3 grid(NTILES_, B_);
  main_kernel<<<grid, 256, 0, stream>>>(x, keep, noise, spat, idx0, idx1, frac,
                                        jit, out_x);
}